// LossGenerator_21225728377303
// MI455X (gfx1250) — compile-verified
//
#include <hip/hip_runtime.h>
#include <cmath>

typedef __attribute__((ext_vector_type(2))) float v2f;
typedef __attribute__((ext_vector_type(8))) float v8f;

#define T_N   11
#define HDIM  2048
#define WDIM  2048
#define HINT  2046
#define WINT  2046
#define NPIX  (HINT * WINT)
#define BLK   256
#define NBC_BLOCKS 80   // 80 * 256 == 10 * 2048 boundary samples

struct MPack { float m[16][12]; };

// ---------------------------------------------------------------------------
// Kernel 1: per-block partial sums of fu^2 over interior pixels.
// D = M (16x12 padded) x Ucenters via V_WMMA_F32_16X16X4_F32.
// ---------------------------------------------------------------------------
__global__ __launch_bounds__(BLK)
void phy_partial_kernel(const float* __restrict__ u, const float* __restrict__ f1,
                        float* __restrict__ partial, MPack mp)
{
  __shared__ float Mlds[16][12];      // padded temporal matrix
  __shared__ float Ulds[8][12][32];   // [wave][s][pix], row 11 = zero K-pad
  __shared__ float Dlds[8][32][20];   // [wave][pix][t], padded stride 20
  __shared__ float red[BLK];

  const int tid  = threadIdx.x;
  const int wave = tid >> 5;
  const int lane = tid & 31;
  const int lo   = lane & 15;
  const int hi   = lane >> 4;

  if (tid < 192) ((float*)Mlds)[tid] = ((const float*)mp.m)[tid];

  long long gidx = (long long)blockIdx.x * BLK + tid;
  const bool valid = gidx < (long long)NPIX;
  const int idc = valid ? (int)gidx : (NPIX - 1);   // clamp: EXEC stays all-1s for WMMA
  const int h = idc / WINT;
  const int w = idc - h * WINT;
  const size_t plane = (size_t)HDIM * WDIM;
  const size_t base  = (size_t)h * WDIM + w;        // top-left of 3x3 window
  const size_t cbase = base + WDIM + 1;             // center

  // Stage center values (the GEMM operand) into registers + LDS.
  float uc[T_N];
  #pragma unroll
  for (int t = 0; t < T_N; ++t) {
    uc[t] = u[(size_t)t * plane + cbase];
    Ulds[wave][t][lane] = uc[t];
  }
  Ulds[wave][11][lane] = 0.0f;
  __syncthreads();

  // D = M x Uc : two 16-pixel column groups per wave, K=12 in 3 chunks of 4.
  #pragma unroll
  for (int g = 0; g < 2; ++g) {
    v8f acc = {};
    #pragma unroll
    for (int c = 0; c < 3; ++c) {
      const int k0 = c * 4 + hi * 2;
      v2f A, B;
      A.x = Mlds[lo][k0];
      A.y = Mlds[lo][k0 + 1];
      B.x = Ulds[wave][k0][g * 16 + lo];
      B.y = Ulds[wave][k0 + 1][g * 16 + lo];
      acc = __builtin_amdgcn_wmma_f32_16x16x4_f32(false, A, false, B,
                                                  (short)0, acc, false, false);
    }
    // lane holds D rows t = r + 8*hi (r=0..7) for pixel g*16+lo
    float* dst = &Dlds[wave][g * 16 + lo][hi * 8];
    #pragma unroll
    for (int r = 0; r < 8; ++r) dst[r] = acc[r];
  }
  __syncthreads();

  // Stencil + residual + squared accumulation (per-lane pixel order restored).
  const float* Dp = &Dlds[wave][lane][0];
  float acc = 0.0f;
  #pragma unroll
  for (int t = 0; t < T_N; ++t) {
    const size_t b = (size_t)t * plane + base;
    const float x00 = u[b],            x01 = u[b + 1],            x02 = u[b + 2];
    const float x10 = u[b + WDIM],                                 x12 = u[b + WDIM + 2];
    const float x20 = u[b + 2*WDIM],   x21 = u[b + 2*WDIM + 1],   x22 = u[b + 2*WDIM + 2];
    const float conv = 0.34520446044393f  * (x00 + x02 + x20 + x22)
                     + 0.309591078922457f * (x01 + x10 + x12 + x21)
                     - 2.619182157203629f * uc[t];
    const float lapk = 0.01f * (conv * 4194304.0f);   // KAPPA * conv / DX^2
    const float fc = f1[(size_t)t * plane + cbase];
    const float fu = Dp[t] - lapk + uc[t] - fc;
    acc = fmaf(fu, fu, acc);
  }
  if (!valid) acc = 0.0f;

  red[tid] = acc;
  __syncthreads();
  for (int s = BLK / 2; s > 0; s >>= 1) {
    if (tid < s) red[tid] += red[tid + s];
    __syncthreads();
  }
  if (tid == 0) partial[blockIdx.x] = red[0];
}

// ---------------------------------------------------------------------------
// Kernel 2: parallel partial sums for the boundary loss (one (t,i) per thread).
// ---------------------------------------------------------------------------
__global__ __launch_bounds__(BLK)
void bc_partial_kernel(const float* __restrict__ u, float* __restrict__ bc_partial)
{
  __shared__ float red[BLK];
  const int tid = threadIdx.x;
  const int j = blockIdx.x * BLK + tid;      // 0 .. 10*2048-1, exact cover
  const int t = j >> 11;                     // / 2048
  const int i = j & (WDIM - 1);

  const float tv = 0.1f + 0.1f * (float)t;
  const float xv = (float)i / 2047.0f;
  const float bv = tv * sqrtf(tv) * sinf(6.283185307179586f * xv);

  const size_t plane = (size_t)HDIM * WDIM;
  const float* ut = u + (size_t)(t + 1) * plane;
  const float cL = ut[(size_t)i * WDIM];                 // left column, row i
  const float cR = ut[(size_t)i * WDIM + (WDIM - 1)];    // right column, row i
  const float rT = ut[i];                                // top row, col i
  const float rB = ut[(size_t)(HDIM - 1) * WDIM + i];    // bottom row, col i
  const float d0 = cL - bv, d1 = cR - bv, d2 = rT - bv, d3 = rB - bv;

  red[tid] = d0 * d0 + d1 * d1 + d2 * d2 + d3 * d3;
  __syncthreads();
  for (int s = BLK / 2; s > 0; s >>= 1) {
    if (tid < s) red[tid] += red[tid + s];
    __syncthreads();
  }
  if (tid == 0) bc_partial[blockIdx.x] = red[0];
}

// ---------------------------------------------------------------------------
// Kernel 3: deterministic final reductions (double).
// ---------------------------------------------------------------------------
__global__ __launch_bounds__(BLK)
void finalize_kernel(const float* __restrict__ partial, int nPartial,
                     const float* __restrict__ bc_partial,
                     float* __restrict__ out)
{
  __shared__ double red[BLK];
  const int tid = threadIdx.x;

  double s = 0.0;
  for (int i = tid; i < nPartial; i += BLK) s += (double)partial[i];
  red[tid] = s;
  __syncthreads();
  for (int k = BLK / 2; k > 0; k >>= 1) { if (tid < k) red[tid] += red[tid + k]; __syncthreads(); }
  if (tid == 0)
    out[0] = (float)(2.0 * red[0] / ((double)T_N * (double)HINT * (double)WINT));
  __syncthreads();

  double sb = (tid < NBC_BLOCKS) ? (double)bc_partial[tid] : 0.0;
  red[tid] = sb;
  __syncthreads();
  for (int k = BLK / 2; k > 0; k >>= 1) { if (tid < k) red[tid] += red[tid + k]; __syncthreads(); }
  if (tid == 0)
    out[1] = (float)(red[0] / (10.0 * 2048.0) / 2048.0);
}

// ---------------------------------------------------------------------------
extern "C" void kernel_launch(void* const* d_in, const int* in_sizes, int n_in,
                              void* d_out, int out_size, void* d_ws, size_t ws_size,
                              hipStream_t stream)
{
  (void)in_sizes; (void)n_in; (void)out_size; (void)ws_size;
  const float* u  = (const float*)d_in[0];
  const float* f1 = (const float*)d_in[1];
  float* out      = (float*)d_out;

  const int nBlocks = (NPIX + BLK - 1) / BLK;
  float* partial    = (float*)d_ws;             // [nBlocks]
  float* bc_partial = partial + nBlocks;        // [NBC_BLOCKS]

  // Build the fractional-derivative matrix (alpha=0.5 -> sqrt weights) in
  // double on the host; pad to 16x12 with zeros for WMMA.
  MPack mp;
  {
    double wgt[10];
    wgt[0] = 1.0;
    for (int j = 1; j <= 9; ++j) wgt[j] = std::sqrt((double)(j + 1)) - std::sqrt((double)j);
    double M[11][11] = {};
    for (int i = 1; i < 11; ++i) {
      M[i][i] = wgt[0];
      M[i][0] -= wgt[i - 1];
      for (int k = 1; k < i; ++k) M[i][k] = -(wgt[i - k - 1] - wgt[i - k]);
    }
    const double pref = std::pow(0.1, -0.5) / 0.886226925452758;
    for (int t = 0; t < 16; ++t)
      for (int sIdx = 0; sIdx < 12; ++sIdx)
        mp.m[t][sIdx] = (t < 11 && sIdx < 11) ? (float)(pref * M[t][sIdx]) : 0.0f;
  }

  phy_partial_kernel<<<nBlocks, BLK, 0, stream>>>(u, f1, partial, mp);
  bc_partial_kernel<<<NBC_BLOCKS, BLK, 0, stream>>>(u, bc_partial);
  finalize_kernel<<<1, BLK, 0, stream>>>(partial, nBlocks, bc_partial, out);
}